// AtomPairProjectionBin_83416854823433
// MI455X (gfx1250) — compile-verified
//
#include <hip/hip_runtime.h>
#include <hip/hip_bf16.h>
#include <math.h>

typedef __attribute__((ext_vector_type(2))) float v2f;
typedef __attribute__((ext_vector_type(8))) float v8f;

#define BATCH 8
#define SEQ   128
#define DATOM 256
#define DHID  32
#define DAB   64      // 2*d_hid
#define K1    21
#define K2    30
#define KTOT  51
#define KPAD  64

// workspace layout (in floats)
#define WS_WPAD 0                               // [32][32][64] packed+padded W1|W2
#define WS_BPAD (32*32*64)                      // [64] packed bias
#define WS_A    (WS_BPAD + 64)                  // a: [B][L][32]
#define WS_BM   (WS_A + BATCH*SEQ*DHID)         // b: [B][L][32]

#define X2OFF (BATCH*SEQ*SEQ*K1)

// ---------------------------------------------------------------------------
// Kernel 0: pack W1 (1024x21) and W2 (1024x30) into Wpad[c][d][64] (k>=51 -> 0)
// and biases into bpad[64].
// ---------------------------------------------------------------------------
__global__ void pack_weights_kernel(const float* __restrict__ W1,
                                    const float* __restrict__ b1,
                                    const float* __restrict__ W2,
                                    const float* __restrict__ b2,
                                    float* __restrict__ ws) {
    int idx = blockIdx.x * blockDim.x + threadIdx.x;
    if (idx < 32 * 32 * KPAD) {
        int c = idx >> 11;          // /2048
        int d = (idx >> 6) & 31;
        int k = idx & 63;
        float v = 0.0f;
        if (k < K1)       v = W1[(c * 32 + d) * K1 + k];
        else if (k < KTOT) v = W2[(c * 32 + d) * K2 + (k - K1)];
        ws[WS_WPAD + idx] = v;
    } else if (idx < 32 * 32 * KPAD + KPAD) {
        int k = idx - 32 * 32 * KPAD;
        float v = 0.0f;
        if (k < K1)       v = b1[k];
        else if (k < KTOT) v = b2[k - K1];
        ws[WS_BPAD + k] = v;
    }
}

// ---------------------------------------------------------------------------
// Kernel 1: per row (b,l): y = gelu_exact(z_row @ W_in + b_in), LayerNorm(64),
// split halves into a / b buffers in workspace.
// One 64-thread block per row; z row staged in LDS for reuse by 64 dots.
// ---------------------------------------------------------------------------
__global__ void __launch_bounds__(64)
proj_gelu_ln_kernel(const float* __restrict__ z,
                    const float* __restrict__ W_in,
                    const float* __restrict__ b_in,
                    const float* __restrict__ ln_g,
                    const float* __restrict__ ln_b,
                    float* __restrict__ ws) {
    __shared__ float sz[DATOM];
    __shared__ float ssum[DAB];
    __shared__ float ssq[DAB];

    const int row = blockIdx.x;            // 0 .. B*L-1
    const int j   = threadIdx.x;           // 0 .. 63

    // stage z row
    const float* zrow = z + (size_t)row * DATOM;
    #pragma unroll
    for (int i = 0; i < DATOM / DAB; ++i)
        sz[j + i * DAB] = zrow[j + i * DAB];
    __syncthreads();

    // dot product: column j of W_in (256x64, row-major)
    float acc = b_in[j];
    #pragma unroll 8
    for (int c = 0; c < DATOM; ++c)
        acc = fmaf(sz[c], W_in[c * DAB + j], acc);

    // exact GELU: 0.5*x*(1+erf(x/sqrt(2)))
    float g = 0.5f * acc * (1.0f + erff(acc * 0.70710678118654752f));

    ssum[j] = g;
    ssq[j]  = g * g;
    __syncthreads();
    #pragma unroll
    for (int s = 32; s > 0; s >>= 1) {
        if (j < s) { ssum[j] += ssum[j + s]; ssq[j] += ssq[j + s]; }
        __syncthreads();
    }
    const float mu  = ssum[0] * (1.0f / DAB);
    const float var = ssq[0] * (1.0f / DAB) - mu * mu;
    const float inv = rsqrtf(var + 1e-5f);

    const float val = (g - mu) * inv * ln_g[j] + ln_b[j];

    if (j < DHID) ws[WS_A  + row * DHID + j]          = val;
    else          ws[WS_BM + row * DHID + (j - DHID)] = val;
}

// ---------------------------------------------------------------------------
// Kernel 2: per (batch b, position l) task:
//   U[d][k]    = sum_c a[b,l,c] * Wpad[c][d][k]          (VALU, into LDS)
//   D[m][k]    = sum_d bmat[b,m,d] * U[d][k]             (WMMA f32 16x16x4)
//   x1[b,l,m,k<21] = D + bias, x2[b,l,m,21<=k<51] = D + bias
// 256 threads = 8 waves; wave w owns M-tile rows [16w,16w+16), 4 N-tiles.
// ---------------------------------------------------------------------------
__global__ void __launch_bounds__(256)
trilinear_wmma_kernel(const float* __restrict__ ws,
                      float* __restrict__ out) {
    __shared__ float sB[SEQ][DHID + 1];   // b[b] rows, padded to kill bank conflicts
    __shared__ float sU[DHID][KPAD];      // U matrix (32 x 64)
    __shared__ float sa[DHID];            // a[b,l,:]

    const int task = blockIdx.x;          // 0 .. B*L-1
    const int bb   = task >> 7;
    const int l    = task & (SEQ - 1);
    const int tid  = threadIdx.x;

    // --- stage b[bb] (128x32) into LDS, coalesced ---
    const float* bmat = ws + WS_BM + (size_t)bb * SEQ * DHID;
    #pragma unroll
    for (int i = tid; i < SEQ * DHID; i += 256) {
        int m = i >> 5, d = i & 31;
        sB[m][d] = bmat[i];
    }
    if (tid < DHID) sa[tid] = ws[WS_A + (size_t)task * DHID + tid];
    __syncthreads();

    // --- U[d][k] = sum_c sa[c] * Wpad[c][d][k] : 8 entries per thread ---
    {
        const int e0 = tid * 8;           // contiguous k-run, fixed d
        const int d  = e0 >> 6;
        const int k0 = e0 & 63;
        float u[8] = {0, 0, 0, 0, 0, 0, 0, 0};
        const float* wp = ws + WS_WPAD + d * KPAD + k0;
        #pragma unroll 4
        for (int c = 0; c < DHID; ++c) {
            const float av = sa[c];
            const float* wrow = wp + c * (32 * KPAD);
            #pragma unroll
            for (int q = 0; q < 8; ++q)
                u[q] = fmaf(av, wrow[q], u[q]);
        }
        #pragma unroll
        for (int q = 0; q < 8; ++q)
            sU[d][k0 + q] = u[q];
    }
    __syncthreads();

    // --- WMMA: D(128x64) = sB(128x32) @ sU(32x64), tiles of 16x16, K-chunks of 4 ---
    const int wave = tid >> 5;
    const int lane = tid & 31;
    const int m0   = wave * 16;
    const int half = (lane < 16) ? 0 : 1;     // K pair / row-half select
    const int l16  = lane & 15;

    const size_t row_base1 = ((size_t)(bb * SEQ + l)) * SEQ;   // * K1 later
    const float* bpad = ws + WS_BPAD;

    #pragma unroll
    for (int nt = 0; nt < 4; ++nt) {
        const int n0 = nt * 16;
        v8f acc = {0.f, 0.f, 0.f, 0.f, 0.f, 0.f, 0.f, 0.f};
        #pragma unroll
        for (int kk = 0; kk < 8; ++kk) {
            const int kb = kk * 4 + half * 2;
            // A fragment: 16x4 f32 (lanes 0-15: K=kb,kb+1 ; lanes 16-31: K=kb+2? no:
            //  half==1 already shifts kb by 2) -> element0 = K=kb, element1 = K=kb+1
            v2f afrag, bfrag;
            afrag.x = sB[m0 + l16][kb];
            afrag.y = sB[m0 + l16][kb + 1];
            // B fragment: 4x16 f32, mirrored striping
            bfrag.x = sU[kb][n0 + l16];
            bfrag.y = sU[kb + 1][n0 + l16];
            acc = __builtin_amdgcn_wmma_f32_16x16x4_f32(
                false, afrag, false, bfrag, (short)0, acc, false, false);
        }

        const int n      = n0 + l16;
        const float bias = bpad[n];
        // D layout: VGPR j -> M = m0 + j (lanes 0-15) or m0 + 8 + j (lanes 16-31)
        #pragma unroll
        for (int j = 0; j < 8; ++j) {
            const int m = m0 + half * 8 + j;
            const float val = acc[j] + bias;
            if (n < K1) {
                out[(row_base1 + m) * K1 + n] = val;
            } else if (n < KTOT) {
                out[X2OFF + (row_base1 + m) * K2 + (n - K1)] = val;
            }
        }
    }
}

// ---------------------------------------------------------------------------
extern "C" void kernel_launch(void* const* d_in, const int* in_sizes, int n_in,
                              void* d_out, int out_size, void* d_ws, size_t ws_size,
                              hipStream_t stream) {
    const float* z    = (const float*)d_in[0];
    const float* W_in = (const float*)d_in[1];
    const float* b_in = (const float*)d_in[2];
    const float* ln_g = (const float*)d_in[3];
    const float* ln_b = (const float*)d_in[4];
    const float* W1   = (const float*)d_in[5];
    const float* b1   = (const float*)d_in[6];
    const float* W2   = (const float*)d_in[7];
    const float* b2   = (const float*)d_in[8];
    float* out = (float*)d_out;
    float* ws  = (float*)d_ws;

    // pack weights/bias (65600 elems)
    pack_weights_kernel<<<(32 * 32 * KPAD + KPAD + 255) / 256, 256, 0, stream>>>(
        W1, b1, W2, b2, ws);

    // stage 1: proj + gelu + layernorm, one block per (b,l) row
    proj_gelu_ln_kernel<<<BATCH * SEQ, 64, 0, stream>>>(
        z, W_in, b_in, ln_g, ln_b, ws);

    // stage 2: trilinear einsum via WMMA, one block per (b,l)
    trilinear_wmma_kernel<<<BATCH * SEQ, 256, 0, stream>>>(ws, out);
}